// HGBlock_730144440555
// MI455X (gfx1250) — compile-verified
//
#include <hip/hip_runtime.h>

typedef __attribute__((ext_vector_type(2))) float v2f;
typedef __attribute__((ext_vector_type(4))) float v4f;
typedef __attribute__((ext_vector_type(8))) float v8f;

// ---------------------------------------------------------------------------
// WMMA helper: D = A(16x4) * B(4x16) + C, fp32, CDNA5 wave32
// A layout (ISA 7.12.2, 32-bit A 16x4): lane L holds row M=L%16,
//   K = 2*(L/16)+vgpr.  B mirrored: lane L holds col N=L%16, K=2*(L/16)+vgpr.
// C/D: lane L, vgpr v -> row M = v + 8*(L/16), col N = L%16.
// ---------------------------------------------------------------------------
static __device__ __forceinline__ v8f wmma_f32(v2f a, v2f b, v8f c) {
  return __builtin_amdgcn_wmma_f32_16x16x4_f32(
      /*neg_a=*/false, a, /*neg_b=*/false, b,
      /*c_mod=*/(short)0, c, /*reuse_a=*/false, /*reuse_b=*/false);
}

#define LDA_PAD 36   // 36*4B = 144B: 16B-aligned b128 stores; 36*r mod 64 hits
                     // all 16 distinct 4-spaced banks -> conflict-free b64 reads
#define LDB_NN 132   // [k][n] stride; 132*4B % 16 == 0; k-rows on disjoint banks

// ---------------------------------------------------------------------------
// NT GEMM: C[M,N] = A[M,K] * W[N,K]^T   (row-major, K contiguous in both)
// 256 threads = 8 wave32; block tile 64x128; wave tile 16x64; K-chunk 32,
// double-buffered through LDS.  Requires M%64==0, N%128==0, K%32==0.
// ---------------------------------------------------------------------------
__global__ void __launch_bounds__(256) gemm_nt_wmma(
    const float* __restrict__ A, const float* __restrict__ W,
    float* __restrict__ C, int M, int N, int K) {
  __shared__ float sA[2][64 * LDA_PAD];    // 18.4 KB
  __shared__ float sB[2][128 * LDA_PAD];   // 36.9 KB
  const int tid = threadIdx.x;
  const int lane = tid & 31;
  const int wave = tid >> 5;
  const int m00 = blockIdx.x * 64;
  const int n00 = blockIdx.y * 128;
  const int mw = (wave & 3) * 16;
  const int nw = (wave >> 2) * 64;
  const int mr = lane & 15;
  const int kb = (lane >> 4) << 1;

  v8f acc[4] = {};
  v4f ra[2], rb[4];

  auto gload = [&](int kk0) {
#pragma unroll
    for (int u = 0; u < 2; ++u) {          // A: 64x32 = 512 float4 chunks
      const int i = tid + u * 256, row = i >> 3, c4 = (i & 7) << 2;
      ra[u] = *(const v4f*)(A + (size_t)(m00 + row) * K + kk0 + c4);
    }
#pragma unroll
    for (int u = 0; u < 4; ++u) {          // B: 128x32 = 1024 float4 chunks
      const int i = tid + u * 256, row = i >> 3, c4 = (i & 7) << 2;
      rb[u] = *(const v4f*)(W + (size_t)(n00 + row) * K + kk0 + c4);
    }
  };
  auto stor = [&](int buf) {
#pragma unroll
    for (int u = 0; u < 2; ++u) {
      const int i = tid + u * 256, row = i >> 3, c4 = (i & 7) << 2;
      *(v4f*)&sA[buf][row * LDA_PAD + c4] = ra[u];
    }
#pragma unroll
    for (int u = 0; u < 4; ++u) {
      const int i = tid + u * 256, row = i >> 3, c4 = (i & 7) << 2;
      *(v4f*)&sB[buf][row * LDA_PAD + c4] = rb[u];
    }
  };

  gload(0);
  stor(0);
  __syncthreads();
  const int ns = K >> 5;
  for (int s = 0; s < ns; ++s) {
    if (s + 1 < ns) gload((s + 1) << 5);   // prefetch next stage (hidden)
    const int buf = s & 1;
#pragma unroll
    for (int k0 = 0; k0 < 32; k0 += 4) {
      const v2f a = *(const v2f*)&sA[buf][(mw + mr) * LDA_PAD + k0 + kb];
#pragma unroll
      for (int t = 0; t < 4; ++t) {
        const v2f b = *(const v2f*)&sB[buf][(nw + t * 16 + mr) * LDA_PAD + k0 + kb];
        acc[t] = wmma_f32(a, b, acc[t]);
      }
    }
    __syncthreads();
    if (s + 1 < ns) stor(buf ^ 1);
    __syncthreads();
  }
  const int rr = (lane >> 4) * 8;
#pragma unroll
  for (int t = 0; t < 4; ++t)
#pragma unroll
    for (int v = 0; v < 8; ++v)
      C[(size_t)(m00 + mw + rr + v) * N + n00 + nw + t * 16 + mr] = acc[t][v];
}

// ---------------------------------------------------------------------------
// Batched NN GEMM with inner relation loop + optional bias:
//   C[z] = sum_{r<nr} A[z*sAb + r*sAr] (MxK, ld K) * B[z*sBb + r*sBr] (KxN, ld N)
// Same tiling; relation loop flattened into the double-buffered stage stream.
// Requires M%64==0, N%128==0, K%32==0.
// ---------------------------------------------------------------------------
__global__ void __launch_bounds__(256) gemm_nn_wmma(
    const float* __restrict__ A, const float* __restrict__ Bm,
    float* __restrict__ C, int M, int N, int K,
    long sAb, long sBb, long sCb, long sAr, long sBr, int nr,
    const float* __restrict__ bias) {
  __shared__ float sA[2][64 * LDA_PAD];    // 18.4 KB
  __shared__ float sB[2][32 * LDB_NN];     // 33.8 KB, layout [k][n]
  const int tid = threadIdx.x;
  const int lane = tid & 31;
  const int wave = tid >> 5;
  const int m00 = blockIdx.x * 64;
  const int n00 = blockIdx.y * 128;
  const int mw = (wave & 3) * 16;
  const int nw = (wave >> 2) * 64;
  const int mr = lane & 15;
  const int kb = (lane >> 4) << 1;
  const float* A0 = A + (long)blockIdx.z * sAb;
  const float* B0 = Bm + (long)blockIdx.z * sBb;
  float* C0 = C + (long)blockIdx.z * sCb;

  v8f acc[4] = {};
  v4f ra[2], rb[4];

  const int spk = K >> 5;                  // stages per relation
  const int ns = nr * spk;                 // total stages

  auto gload = [&](int ss) {
    const int r = ss / spk, kk0 = (ss % spk) << 5;
    const float* Ar = A0 + (long)r * sAr;
    const float* Br = B0 + (long)r * sBr;
#pragma unroll
    for (int u = 0; u < 2; ++u) {          // A: 64x32
      const int i = tid + u * 256, row = i >> 3, c4 = (i & 7) << 2;
      ra[u] = *(const v4f*)(Ar + (size_t)(m00 + row) * K + kk0 + c4);
    }
#pragma unroll
    for (int u = 0; u < 4; ++u) {          // B: 32x128, coalesced along n
      const int i = tid + u * 256, row = i >> 5, c4 = (i & 31) << 2;
      rb[u] = *(const v4f*)(Br + (size_t)(kk0 + row) * N + n00 + c4);
    }
  };
  auto stor = [&](int buf) {
#pragma unroll
    for (int u = 0; u < 2; ++u) {
      const int i = tid + u * 256, row = i >> 3, c4 = (i & 7) << 2;
      *(v4f*)&sA[buf][row * LDA_PAD + c4] = ra[u];
    }
#pragma unroll
    for (int u = 0; u < 4; ++u) {
      const int i = tid + u * 256, row = i >> 5, c4 = (i & 31) << 2;
      *(v4f*)&sB[buf][row * LDB_NN + c4] = rb[u];
    }
  };

  gload(0);
  stor(0);
  __syncthreads();
  for (int ss = 0; ss < ns; ++ss) {
    if (ss + 1 < ns) gload(ss + 1);
    const int buf = ss & 1;
#pragma unroll
    for (int k0 = 0; k0 < 32; k0 += 4) {
      const v2f a = *(const v2f*)&sA[buf][(mw + mr) * LDA_PAD + k0 + kb];
#pragma unroll
      for (int t = 0; t < 4; ++t) {
        const int n = nw + t * 16 + mr;
        v2f b;
        b.x = sB[buf][(k0 + kb) * LDB_NN + n];
        b.y = sB[buf][(k0 + kb + 1) * LDB_NN + n];
        acc[t] = wmma_f32(a, b, acc[t]);
      }
    }
    __syncthreads();
    if (ss + 1 < ns) stor(buf ^ 1);
    __syncthreads();
  }
  const int rr = (lane >> 4) * 8;
#pragma unroll
  for (int t = 0; t < 4; ++t)
#pragma unroll
    for (int v = 0; v < 8; ++v) {
      float o = acc[t][v];
      if (bias) o += bias[m00 + mw + rr + v];
      C0[(size_t)(m00 + mw + rr + v) * N + n00 + nw + t * 16 + mr] = o;
    }
}

// ---------------------------------------------------------------------------
// ga[b,r,g,h] = sum over adj[r,n,m] with label(m)=g, label(n)=h.
// One workgroup per (b,r); 256KB LDS accumulator + 4KB labels (dynamic LDS).
// ---------------------------------------------------------------------------
__global__ void __launch_bounds__(256) ga_kernel(
    const float* __restrict__ adj, const int* __restrict__ label,
    float* __restrict__ ga) {
  extern __shared__ float sm[];
  float* gal = sm;                    // 65536 floats
  int* lab = (int*)(sm + 65536);      // 1024 ints
  const int b = blockIdx.x / 9;
  const int r = blockIdx.x % 9;
  for (int i = threadIdx.x; i < 65536; i += 256) gal[i] = 0.f;
  for (int i = threadIdx.x; i < 1024; i += 256) lab[i] = label[b * 1024 + i];
  __syncthreads();
  const float* ar = adj + ((size_t)r << 20);
  for (size_t i = threadIdx.x; i < (size_t)1024 * 1024; i += 256) {
    const int n = (int)(i >> 10), m = (int)(i & 1023);
    atomicAdd(&gal[lab[m] * 256 + lab[n]], ar[i]);
  }
  __syncthreads();
  float* go = ga + (size_t)blockIdx.x * 65536;
  for (int i = threadIdx.x; i < 65536; i += 256) go[i] = gal[i];
}

// coef[b,r,g,:] = (r==3 ? ga : relu(ga[b]-ga[7-b])) / (sum_h ga[b,r,g,h] + 1)
__global__ void __launch_bounds__(256) coef_kernel(
    const float* __restrict__ ga, float* __restrict__ coef) {
  const int g = blockIdx.x & 255;
  const int r = (blockIdx.x >> 8) % 9;
  const int b = blockIdx.x / (9 * 256);
  const size_t row = ((size_t)(b * 9 + r) * 256 + g) * 256;
  __shared__ float red[256];
  const float v = ga[row + threadIdx.x];
  red[threadIdx.x] = v;
  __syncthreads();
  for (int s = 128; s > 0; s >>= 1) {
    if (threadIdx.x < s) red[threadIdx.x] += red[threadIdx.x + s];
    __syncthreads();
  }
  const float den = red[0] + 1.0f;
  float num;
  if (r == 3) num = v;
  else {
    const size_t rowf = ((size_t)((7 - b) * 9 + r) * 256 + g) * 256;
    num = fmaxf(v - ga[rowf + threadIdx.x], 0.f);
  }
  coef[row + threadIdx.x] = num / den;
}

// cnt[b,g] = #cells with label g
__global__ void __launch_bounds__(256) count_kernel(
    const int* __restrict__ label, float* __restrict__ cnt) {
  __shared__ int c[256];
  c[threadIdx.x] = 0;
  __syncthreads();
  for (int n = threadIdx.x; n < 1024; n += 256)
    atomicAdd(&c[label[blockIdx.x * 1024 + n]], 1);
  __syncthreads();
  cnt[blockIdx.x * 256 + threadIdx.x] = (float)c[threadIdx.x];
}

// x[b,g,c] = (1/(1+eps)) * sum_{n:label=g} inp[b,c,n]
// grid (B, C/128), 128 threads (one channel each), 128KB dynamic LDS buckets.
__global__ void __launch_bounds__(128) pool_kernel(
    const float* __restrict__ inp, const int* __restrict__ label,
    float* __restrict__ x) {
  extern __shared__ float pm[];            // 256*128 floats
  __shared__ int lab[1024];
  const int b = blockIdx.x;
  const int c = blockIdx.y * 128 + threadIdx.x;
  for (int i = threadIdx.x; i < 1024; i += 128) lab[i] = label[b * 1024 + i];
  for (int i = threadIdx.x; i < 256 * 128; i += 128) pm[i] = 0.f;
  __syncthreads();
  const float* ir = inp + ((size_t)b * 512 + c) * 1024;
  for (int n = 0; n < 1024; ++n)
    pm[lab[n] * 128 + threadIdx.x] += ir[n];   // distinct lane -> distinct cell
  __syncthreads();
  const float inv = 1.0f / (1.0f + 1e-7f);
  for (int g = 0; g < 256; ++g)
    x[((size_t)b * 256 + g) * 512 + c] = pm[g * 128 + threadIdx.x] * inv;
}

// BatchNorm1d(G) stats over (B,C) per group: stats[g]=mean, stats[256+g]=rsqrt(var+eps)
__global__ void __launch_bounds__(256) bn_stats_kernel(
    const float* __restrict__ y, float* __restrict__ stats) {
  const int g = blockIdx.x;
  __shared__ float s1[256], s2[256];
  float sum = 0.f, ssq = 0.f;
  for (int i = threadIdx.x; i < 8 * 512; i += 256) {
    const int b = i >> 9, c = i & 511;
    const float v = y[((size_t)b * 256 + g) * 512 + c];
    sum += v; ssq += v * v;
  }
  s1[threadIdx.x] = sum; s2[threadIdx.x] = ssq;
  __syncthreads();
  for (int s = 128; s > 0; s >>= 1) {
    if (threadIdx.x < s) { s1[threadIdx.x] += s1[threadIdx.x + s];
                           s2[threadIdx.x] += s2[threadIdx.x + s]; }
    __syncthreads();
  }
  if (threadIdx.x == 0) {
    const float mean = s1[0] * (1.f / 4096.f);
    const float var = s2[0] * (1.f / 4096.f) - mean * mean;
    stats[g] = mean;
    stats[256 + g] = rsqrtf(var + 1e-5f);
  }
}

// out = gamma*(y-mean)*invstd + beta (+resid) (relu?)   over (B,G,C)
__global__ void __launch_bounds__(256) bn_apply_kernel(
    const float* __restrict__ y, float* __restrict__ out,
    const float* __restrict__ stats, const float* __restrict__ gamma,
    const float* __restrict__ beta, const float* __restrict__ resid, int relu) {
  const size_t i = (size_t)blockIdx.x * 256 + threadIdx.x;
  const int g = (int)((i >> 9) & 255);
  float v = (y[i] - stats[g]) * stats[256 + g] * gamma[g] + beta[g];
  if (resid) v += resid[i];
  if (relu) v = fmaxf(v, 0.f);
  out[i] = v;
}

// wsum[j,c] = sum_r wg[j, r*512 + c]   (folds tile(h,9) @ wg.T)
__global__ void __launch_bounds__(256) wsum_kernel(
    const float* __restrict__ wg, float* __restrict__ wsum) {
  const size_t i = (size_t)blockIdx.x * 256 + threadIdx.x;  // 4608*512
  const int j = (int)(i >> 9), c = (int)(i & 511);
  float s = 0.f;
#pragma unroll
  for (int r = 0; r < 9; ++r) s += wg[(size_t)j * 4608 + r * 512 + c];
  wsum[i] = s;
}

// Unpool (gather + 1/(cnt+eps)), torch raw (B,N,C)->(B,C,H,W) reinterpret,
// concat with inp into xcat (B, 1024, 1024) row-major (row=channel, col=hw).
__global__ void __launch_bounds__(256) xcat_kernel(
    const float* __restrict__ xfin, const float* __restrict__ cnt,
    const int* __restrict__ label, const float* __restrict__ inp,
    float* __restrict__ xcat) {
  const size_t i = (size_t)blockIdx.x * 256 + threadIdx.x;  // B*1024*1024
  const int hw = (int)(i & 1023);
  const size_t t = i >> 10;
  const int c = (int)(t & 1023);
  const int b = (int)(t >> 10);
  float v;
  if (c < 512) {
    const int L = c * 1024 + hw;          // flat index in (C,H,W) view
    const int n = L >> 9, cc = L & 511;   // maps to (n,c) of (N,C) buffer
    const int g = label[b * 1024 + n];
    v = xfin[((size_t)b * 256 + g) * 512 + cc] / (cnt[b * 256 + g] + 1e-7f);
  } else {
    v = inp[((size_t)b * 512 + (c - 512)) * 1024 + hw];
  }
  xcat[i] = v;
}

// ---------------------------------------------------------------------------
extern "C" void kernel_launch(void* const* d_in, const int* in_sizes, int n_in,
                              void* d_out, int out_size, void* d_ws, size_t ws_size,
                              hipStream_t stream) {
  const float* inp   = (const float*)d_in[0];
  const int*   label = (const int*)d_in[1];
  const float* adj   = (const float*)d_in[2];
  const float* w1    = (const float*)d_in[3];
  const float* wg    = (const float*)d_in[4];
  const float* w2    = (const float*)d_in[5];
  const float* gam   = (const float*)d_in[6];
  const float* bet   = (const float*)d_in[7];
  const float* cw    = (const float*)d_in[8];
  const float* cbias = (const float*)d_in[9];
  float* out = (float*)d_out;
  float* ws  = (float*)d_ws;

  float* ga    = ws;                  // 4,718,592 f
  float* coef  = ga + 4718592;        // 4,718,592 f
  float* cnt   = coef + 4718592;      // 2,048 f
  float* x     = cnt + 2048;          // 1,048,576 f
  float* h     = x + 1048576;         // 1,048,576 f
  float* h2    = h + 1048576;         // 1,048,576 f
  float* stats = h2 + 1048576;        // 512 f
  float* wsum  = stats + 512;         // 2,359,296 f
  float* e     = wsum + 2359296;      // 9,437,184 f  (total ~97.5 MB)
  float* xcat  = ws;                  // aliases ga/coef (dead after blocks)

  // coefficient matrix + pooling
  ga_kernel<<<dim3(72), dim3(256), 266240, stream>>>(adj, label, ga);
  coef_kernel<<<dim3(8 * 9 * 256), dim3(256), 0, stream>>>(ga, coef);
  count_kernel<<<dim3(8), dim3(256), 0, stream>>>(label, cnt);
  pool_kernel<<<dim3(8, 4), dim3(128), 131072, stream>>>(inp, label, x);

  // three residual blocks
  for (int i = 0; i < 3; ++i) {
    const float* w1i = w1 + (size_t)i * 512 * 512;
    const float* w2i = w2 + (size_t)i * 512 * 512;
    const float* wgi = wg + (size_t)i * 4608 * 4608;
    const float* g0 = gam + (size_t)(i * 3 + 0) * 256, *b0 = bet + (size_t)(i * 3 + 0) * 256;
    const float* g1 = gam + (size_t)(i * 3 + 1) * 256, *b1 = bet + (size_t)(i * 3 + 1) * 256;
    const float* g2 = gam + (size_t)(i * 3 + 2) * 256, *b2 = bet + (size_t)(i * 3 + 2) * 256;

    // h = x @ w1.T
    gemm_nt_wmma<<<dim3(32, 4), 256, 0, stream>>>(x, w1i, h, 2048, 512, 512);
    bn_stats_kernel<<<256, 256, 0, stream>>>(h, stats);
    bn_apply_kernel<<<4096, 256, 0, stream>>>(h, h, stats, g0, b0, nullptr, 1);
    // e = h @ wsum.T  (graph conv folded over relations)
    wsum_kernel<<<9216, 256, 0, stream>>>(wgi, wsum);
    gemm_nt_wmma<<<dim3(32, 36), 256, 0, stream>>>(h, wsum, e, 2048, 4608, 512);
    // h2 = sum_r coef[b,r] @ e[b,r]
    gemm_nn_wmma<<<dim3(4, 4, 8), 256, 0, stream>>>(
        coef, e, h2, 256, 512, 256,
        (long)9 * 65536, (long)9 * 131072, (long)131072,
        (long)65536, (long)131072, 9, nullptr);
    bn_stats_kernel<<<256, 256, 0, stream>>>(h2, stats);
    bn_apply_kernel<<<4096, 256, 0, stream>>>(h2, h2, stats, g1, b1, nullptr, 1);
    // h = h2 @ w2.T ; x = relu(bn(h) + x)
    gemm_nt_wmma<<<dim3(32, 4), 256, 0, stream>>>(h2, w2i, h, 2048, 512, 512);
    bn_stats_kernel<<<256, 256, 0, stream>>>(h, stats);
    bn_apply_kernel<<<4096, 256, 0, stream>>>(h, x, stats, g2, b2, x, 1);
  }

  // unpool + concat, then out = conv_w @ xcat + conv_b  (per-batch NN GEMM)
  xcat_kernel<<<dim3(32768), dim3(256), 0, stream>>>(x, cnt, label, inp, xcat);
  gemm_nn_wmma<<<dim3(8, 8, 8), 256, 0, stream>>>(
      cw, xcat, out, 512, 1024, 1024,
      0L, (long)1024 * 1024, (long)512 * 1024, 0L, 0L, 1, cbias);
}